// PermutationRandom_12738873000451
// MI455X (gfx1250) — compile-verified
//
#include <hip/hip_runtime.h>
#include <hip/hip_bf16.h>
#include <stdint.h>

// Problem shape (fixed by the reference)
#define BB 16
#define LL 4096
#define DD 1024
#define ROWS_PER_BLK 16   // rows gathered by ONE TDM gather-mode descriptor

typedef __attribute__((ext_vector_type(4))) unsigned int uint32x4_v;
typedef __attribute__((ext_vector_type(8))) int          int32x8_v;
typedef __attribute__((ext_vector_type(4))) int          int32x4_v;

// ---------------------------------------------------------------------------
// Counter-based hash -> uniform keys (stands in for jax.random.uniform; any
// uniform key stream yields a valid random permutation of mask==1 tokens).
// ---------------------------------------------------------------------------
__device__ __forceinline__ uint32_t hash_u32(uint32_t x) {
  x ^= x >> 16; x *= 0x7feb352dU;
  x ^= x >> 15; x *= 0x846ca68bU;
  x ^= x >> 16;
  return x;
}

// ---------------------------------------------------------------------------
// Kernel 1: per-batch key generation + bitonic argsort of 4096 (key, idx)
// pairs in LDS. 16 blocks x 1024 threads (32 waves), 32 KB of 320 KB/WGP LDS.
// perm written as int32 to workspace (consumed by kernel 2) and as float to
// the tail of d_out (second tuple output).
// ---------------------------------------------------------------------------
__global__ void __launch_bounds__(1024)
perm_sort_kernel(const int* __restrict__ mask,
                 int* __restrict__ perm_i,
                 float* __restrict__ perm_f) {
  __shared__ float skey[LL];
  __shared__ int   sidx[LL];

  const int b = blockIdx.x;

  for (int i = threadIdx.x; i < LL; i += blockDim.x) {
    const int m = mask[b * LL + i];
    const uint32_t h = hash_u32(0x9E3779B9u * (uint32_t)(b * LL + i) + 42u);
    const float u    = (float)(h >> 8) * (1.0f / 16777216.0f);   // [0, 1)
    const float padk = 1.0f + (float)i * (1.0f / (float)LL);     // >= 1.0, ordered
    skey[i] = (m == 1) ? u : padk;
    sidx[i] = i;
  }
  __syncthreads();

  // Bitonic sort ascending by key, carrying the index.
  for (int k = 2; k <= LL; k <<= 1) {
    for (int j = k >> 1; j > 0; j >>= 1) {
      for (int i = threadIdx.x; i < LL; i += blockDim.x) {
        const int p = i ^ j;
        if (p > i) {
          const bool up = ((i & k) == 0);
          const float ka = skey[i];
          const float kc = skey[p];
          if (up ? (ka > kc) : (ka < kc)) {
            skey[i] = kc; skey[p] = ka;
            const int t = sidx[i]; sidx[i] = sidx[p]; sidx[p] = t;
          }
        }
      }
      __syncthreads();
    }
  }

  for (int i = threadIdx.x; i < LL; i += blockDim.x) {
    const int v = sidx[i];
    perm_i[b * LL + i] = v;
    perm_f[b * LL + i] = (float)v;   // indices <= 4095 are exact in f32
  }
}

// ---------------------------------------------------------------------------
// Kernel 2: one wave per 16 output rows. A single TDM *gather-mode* load
// (ISA §8.7: gather_mode=1, 16-bit row indices in descriptor groups 2/3,
// tile_dim1 = #indices) pulls 16 arbitrary rows x[b, perm[...], :] (64 KB)
// into LDS in one instruction; one contiguous 1-D tensor_store_from_lds
// (tile_dim0 = 16384 elements) streams them to out. TENSORcnt tracks both.
// 4096 single-wave blocks keep the 23.3 TB/s HBM interface saturated while
// VALU stays essentially idle.
// ---------------------------------------------------------------------------
__global__ void __launch_bounds__(32)
gather_tdm_kernel(const float* __restrict__ x,
                  const int* __restrict__ perm_i,
                  float* __restrict__ out) {
  __shared__ float lbuf[ROWS_PER_BLK * DD];   // 64 KB staging tile

  const int row0 = blockIdx.x * ROWS_PER_BLK; // first output row (never crosses batch)
  const int b    = row0 >> 12;                // row0 / LL
  const uint64_t src_base = (uint64_t)(uintptr_t)(x + (size_t)b * LL * DD);
  const uint64_t dst_ga   = (uint64_t)(uintptr_t)(out + (size_t)row0 * DD);
  const uint32_t lds_addr = (uint32_t)(uintptr_t)(&lbuf[0]);

  // Pack the 16 gathered row indices (< 4096, so 16-bit) into groups 2 and 3:
  // 16-bit index mode -> g2 = idx0..7 (two per dword), g3 = idx8..15.
  int32x4_v g2, g3;
#pragma unroll
  for (int w = 0; w < 4; ++w) {
    const int lo2 = perm_i[row0 + 2 * w];
    const int hi2 = perm_i[row0 + 2 * w + 1];
    g2[w] = (lo2 & 0xFFFF) | (hi2 << 16);
    const int lo3 = perm_i[row0 + 8 + 2 * w];
    const int hi3 = perm_i[row0 + 8 + 2 * w + 1];
    g3[w] = (lo3 & 0xFFFF) | (hi3 << 16);
  }

  // ---- gather-load descriptor: 16 rows of 1024 f32 from a 4096x1024 tensor
  uint32x4_v g0l;
  g0l.x = 0x80000001u;                            // count=1 | gather_mode (bit31), 16-bit idx (bit30=0)
  g0l.y = lds_addr;                               // LDS byte address
  g0l.z = (uint32_t)src_base;                     // global_addr[31:0] (batch base)
  g0l.w = (uint32_t)(src_base >> 32) | (2u << 30);// global_addr[56:32] | type=2
  int32x8_v g1l;
  g1l[0] = 0x00020000;           // workgroup_mask=0, data_size=4B
  g1l[1] = (int)(DD << 16);      // tensor_dim0 = 1024 (bits 79:48, low half)
  g1l[2] = (int)(LL << 16);      // tensor_dim1 = 4096 rows (bits 111:80, low half)
  g1l[3] = (int)(DD << 16);      // tile_dim0 = 1024 (row width, bits 127:112)
  g1l[4] = ROWS_PER_BLK;         // tile_dim1 = #valid gather indices = 16 (bits 143:128)
  g1l[5] = DD;                   // tensor_dim0_stride = 1024 (bits 207:160, low 32)
  g1l[6] = 0;
  g1l[7] = 0;

  // ---- contiguous store descriptor: 1-D tile of 16*1024 = 16384 elements
  const int N = ROWS_PER_BLK * DD;                // 16384, fits 16-bit tile_dim0
  uint32x4_v g0s;
  g0s.x = 1u;                                     // count=1, normal mode
  g0s.y = lds_addr;
  g0s.z = (uint32_t)dst_ga;
  g0s.w = (uint32_t)(dst_ga >> 32) | (2u << 30);
  int32x8_v g1s;
  g1s[0] = 0x00020000;                            // data_size=4B
  g1s[1] = (int)((N & 0xFFFF) << 16);             // tensor_dim0 = 16384 (low half)
  g1s[2] = (int)(((unsigned)N >> 16) | (1u << 16)); // tensor_dim0 hi | tensor_dim1 = 1
  g1s[3] = (int)(N << 16);                        // tile_dim0 = 16384
  g1s[4] = 0;                                     // tile_dim1/2 unused
  g1s[5] = N;                                     // tensor_dim0_stride = 16384
  g1s[6] = 0;
  g1s[7] = 0;

  const int32x4_v gz4 = {0, 0, 0, 0};
  const int32x8_v gz8 = {0, 0, 0, 0, 0, 0, 0, 0};

  // DMA: gather 16 permuted rows -> LDS
  __builtin_amdgcn_tensor_load_to_lds(g0l, g1l, g2, g3, gz8, 0);
  __builtin_amdgcn_s_wait_tensorcnt(0);
  // DMA: LDS -> contiguous 64 KB destination block
  __builtin_amdgcn_tensor_store_from_lds(g0s, g1s, gz4, gz4, gz8, 0);
  __builtin_amdgcn_s_wait_tensorcnt(0);
}

// ---------------------------------------------------------------------------
// Launch: inputs are x (f32, B*L*D) and mask (i32, B*L); d_out is
// [x_perm (B*L*D f32) | perm (B*L, stored as float values)].
// d_ws holds the int32 permutation (256 KB) consumed by the gather kernel.
// ---------------------------------------------------------------------------
extern "C" void kernel_launch(void* const* d_in, const int* in_sizes, int n_in,
                              void* d_out, int out_size, void* d_ws, size_t ws_size,
                              hipStream_t stream) {
  const float* x    = (const float*)d_in[0];
  const int*   mask = (const int*)d_in[1];
  float* out    = (float*)d_out;
  float* perm_f = out + (size_t)BB * LL * DD;   // tuple output #2
  int*   perm_i = (int*)d_ws;                   // scratch: B*L int32 = 256 KB

  perm_sort_kernel<<<BB, 1024, 0, stream>>>(mask, perm_i, perm_f);
  gather_tdm_kernel<<<(BB * LL) / ROWS_PER_BLK, 32, 0, stream>>>(x, perm_i, out);
}